// FieldFactorizeLayer_28252294873091
// MI455X (gfx1250) — compile-verified
//
#include <hip/hip_runtime.h>
#include <stdint.h>

#define N_FIELDS 8
#define N_NS     4
#define DIM_NS   100000
#define DIM_SEQ  1000
#define SEQ_LEN  50
#define DIM      64
#define BATCH    16384
#define N_SLOTS  7   // N_FIELDS - 1

typedef __attribute__((ext_vector_type(2))) float v2f;
typedef __attribute__((ext_vector_type(8))) float v8f;

// ---------------------------------------------------------------------------
// Kernel 1: mean-pool the sequence embeddings.
// grid = (28 tables, 8 row-chunks), block = 256 threads (8 waves, wave32).
// The 1000x64 f32 table (250 KB) is staged into LDS via CDNA5 async
// global->LDS copies, then each wave gathers/pools rows out of LDS.
// pooled layout: pooled[(f*7+s)*BATCH*DIM + row*DIM + d]
// ---------------------------------------------------------------------------
__global__ void pool_seq_kernel(const float* __restrict__ tables_seq,
                                const int* __restrict__ idx_seq,
                                float* __restrict__ pooled) {
  extern __shared__ float lds[];
  const int tid  = threadIdx.x;
  const int lane = tid & 31;
  const int wave = tid >> 5;
  const int fs   = blockIdx.x;        // table id = f*N_SLOTS + slot
  const int f    = fs / N_SLOTS;      // owning sequence field (0..3)

  // ---- Stage table fs (DIM_SEQ*DIM floats = 250 KB) into LDS (async DMA) ----
  const float* gtab = tables_seq + (size_t)fs * DIM_SEQ * DIM;
  const uint32_t lds_base = (uint32_t)(uintptr_t)lds;  // low 32 bits = LDS byte addr
  const int n_pkts = DIM_SEQ * DIM / 4;                // 16000 x 16B packets
  for (int i = tid; i < n_pkts; i += 256) {
    uint32_t ldsoff = lds_base + (uint32_t)i * 16u;
    uint32_t goff   = (uint32_t)i * 16u;
    // GVS mode: mem = SGPR base + 32-bit VGPR byte offset; dest = LDS byte addr
    asm volatile("global_load_async_to_lds_b128 %0, %1, %2"
                 :: "v"(ldsoff), "v"(goff), "s"(gtab)
                 : "memory");
  }
  asm volatile("s_wait_asynccnt 0" ::: "memory");
  __syncthreads();

  // ---- Pool 2048 rows per block; each wave handles one row at a time,
  //      lane l covers dims {2l, 2l+1} (ds_load_b64 gathers from LDS). ----
  const int rowStart = blockIdx.y * (BATCH / 8);
  const int rowEnd   = rowStart + (BATCH / 8);
  float* pout = pooled + (size_t)fs * BATCH * DIM;

  for (int r = rowStart + wave; r < rowEnd; r += 8) {
    const int* ip = idx_seq + ((size_t)f * BATCH + r) * SEQ_LEN;
    v2f acc = {0.0f, 0.0f};
#pragma unroll 5
    for (int t = 0; t < SEQ_LEN; ++t) {
      const int v = ip[t];
      const v2f* lrow = (const v2f*)(lds + v * DIM);
      acc += lrow[lane];
    }
    acc *= (1.0f / (float)SEQ_LEN);
    *(v2f*)(pout + (size_t)r * DIM + 2 * lane) = acc;
  }
}

// ---------------------------------------------------------------------------
// Kernel 2: sum over 28 field pairs of <ea, eb> per row, via WMMA diag trick.
// One wave handles 16 rows: per pair, A = U[16 rows x 4 dims],
// B = V[4 dims x 16 rows], accumulate C (16x16) over K=64; the diagonal of C
// accumulates the per-row dot products. 448 v_wmma_f32_16x16x4_f32 per wave.
// ---------------------------------------------------------------------------
__device__ __forceinline__ const float* emb_ptr(int field, int other, int row,
                                                const float* __restrict__ tns,
                                                const float* __restrict__ pooled,
                                                const int* __restrict__ idx_ns) {
  const int slot = (other < field) ? other : other - 1;
  if (field < N_NS) {
    const int idx = idx_ns[field * BATCH + row];   // idx_ns is (4, B, 1)
    return tns + ((size_t)(field * N_SLOTS + slot) * DIM_NS + (size_t)idx) * DIM;
  }
  return pooled + (((size_t)((field - N_NS) * N_SLOTS + slot) * BATCH) + (size_t)row) * DIM;
}

__global__ void pair_dot_kernel(const float* __restrict__ tables_ns,
                                const int* __restrict__ idx_ns,
                                const float* __restrict__ pooled,
                                float* __restrict__ out) {
  const int lane = threadIdx.x & 31;
  const int wave = threadIdx.x >> 5;
  const int g    = blockIdx.x * 8 + wave;  // 1024 groups of 16 rows
  const int rowbase = g * 16;
  const int m    = lane & 15;              // this lane's row within the group
  const int row  = rowbase + m;
  const int klo  = (lane >= 16) ? 2 : 0;   // A/B f32 layout: lanes 16-31 hold K+2/K+3

  v8f c0 = {};
  v8f c1 = {};

  for (int a = 0; a < N_FIELDS; ++a) {
    for (int b = a + 1; b < N_FIELDS; ++b) {
      const float* pa = emb_ptr(a, b, row, tables_ns, pooled, idx_ns);
      const float* pb = emb_ptr(b, a, row, tables_ns, pooled, idx_ns);
#pragma unroll
      for (int k = 0; k < DIM; k += 8) {
        v2f av0 = *(const v2f*)(pa + k + klo);
        v2f bv0 = *(const v2f*)(pb + k + klo);
        c0 = __builtin_amdgcn_wmma_f32_16x16x4_f32(
            /*neg_a=*/false, av0, /*neg_b=*/false, bv0,
            /*c_mod=*/(short)0, c0, /*reuse_a=*/false, /*reuse_b=*/false);
        v2f av1 = *(const v2f*)(pa + k + 4 + klo);
        v2f bv1 = *(const v2f*)(pb + k + 4 + klo);
        c1 = __builtin_amdgcn_wmma_f32_16x16x4_f32(
            false, av1, false, bv1, (short)0, c1, false, false);
      }
    }
  }

  // c = c0 + c1, then extract the diagonal of the 16x16 accumulator.
  // C layout: VGPR r, lanes 0-15 -> (M=r, N=lane); lanes 16-31 -> (M=r+8, N=lane-16).
  // Diag (r,r) lives in lane r of c[r]; diag (r+8,r+8) lives in lane r+24 of c[r].
  float diag = 0.0f;
#pragma unroll
  for (int r = 0; r < 8; ++r) {
    float cr = c0[r] + c1[r];
    float t0 = __shfl(cr, r, 32);
    float t1 = __shfl(cr, r + 24, 32);
    if (m == r)     diag = t0;
    if (m == r + 8) diag = t1;
  }
  if (lane < 16) out[rowbase + lane] = diag;
}

// ---------------------------------------------------------------------------
extern "C" void kernel_launch(void* const* d_in, const int* in_sizes, int n_in,
                              void* d_out, int out_size, void* d_ws, size_t ws_size,
                              hipStream_t stream) {
  const float* tables_ns  = (const float*)d_in[0];  // (28, 100000, 64) f32
  const float* tables_seq = (const float*)d_in[1];  // (28, 1000, 64) f32
  const int*   idx_ns     = (const int*)d_in[2];    // (4, B, 1) i32
  const int*   idx_seq    = (const int*)d_in[3];    // (4, B, 50) i32
  float*       out        = (float*)d_out;          // (B, 1) f32
  float*       pooled     = (float*)d_ws;           // 28 * B * 64 f32 = 112 MiB

  // Kernel 1: 28 tables x 8 row-chunks, 250 KB dynamic LDS per block.
  dim3 g1(N_NS * N_SLOTS, 8);
  pool_seq_kernel<<<g1, 256, DIM_SEQ * DIM * sizeof(float), stream>>>(
      tables_seq, idx_seq, pooled);

  // Kernel 2: 1024 row-groups of 16 rows, 8 waves per block.
  pair_dot_kernel<<<BATCH / (16 * 8), 256, 0, stream>>>(
      tables_ns, idx_ns, pooled, out);
}